// Consistent_loss_up_3_25288767439316
// MI455X (gfx1250) — compile-verified
//
#include <hip/hip_runtime.h>
#include <hip/hip_bf16.h>

typedef __attribute__((ext_vector_type(2))) float v2f;
typedef __attribute__((ext_vector_type(8))) float v8f;

#define H 256
#define W 256

// One block per input column j (== output row j of the scattered buffers).
// Threads 0..255 cover rows i (phase 1: scatter-max into LDS) and then
// output columns c (phase 2: masked |diff| + tree reduction).
__global__ __launch_bounds__(256) void scatter_loss_kernel(
    const float* __restrict__ up,
    const float* __restrict__ left,
    const float* __restrict__ right,
    float* __restrict__ partial) {
  __shared__ unsigned sh_l[W];   // scatter-max targets (float bits, all >= 0)
  __shared__ unsigned sh_r[W];
  __shared__ float red[256];

  const int j   = blockIdx.x;    // input column / output row
  const int tid = threadIdx.x;

  sh_l[tid] = 0u;
  sh_r[tid] = 0u;
  __syncthreads();

  // ---- phase 1: scatter-max of val_l/val_r into bucket(up) ----
  {
    const int i = tid;                     // row index
    const float u = up[i * W + j];         // column read; whole array lives in L2
    // bucket = round_half_even(u*50 + 110); avoid FMA contraction to match XLA
    const float t = __fadd_rn(__fmul_rn(u, 50.0f), 110.0f);
    const int bucket = __float2int_rn(t);  // in [110,160] for u in [0,1)
    const bool m = (u >= 0.0235f);
    const float val_r = (m && (i >  128)) ? (float)(i - 128) * (1.0f / 60.0f) : 0.0f;
    const float val_l = (m && (i <= 128)) ? (float)(128 - i) * (1.0f / 60.0f) : 0.0f;
    // non-negative floats: unsigned max == float max; zero scatter is a no-op
    if (val_r > 0.0f) atomicMax(&sh_r[bucket], __float_as_uint(val_r));
    if (val_l > 0.0f) atomicMax(&sh_l[bucket], __float_as_uint(val_l));
  }
  __syncthreads();

  // ---- phase 2: masked |diff| contributions for output row j ----
  {
    const int c = tid;
    const float up2l = __uint_as_float(sh_l[c]);
    const float up2r = __uint_as_float(sh_r[c]);
    const float l = left[j * W + c];    // coalesced row reads
    const float r = right[j * W + c];
    const float dl = fabsf(up2l - l);
    const float dr = fabsf(up2r - r);
    const float ml = ((dl < 0.2f) && (up2l != 0.0f)) ? dl : 0.0f;
    const float mr = ((dr < 0.2f) && (up2r != 0.0f)) ? dr : 0.0f;
    red[tid] = ml + mr;
  }
  __syncthreads();

  // deterministic tree reduction of 256 contributions
  for (int s = 128; s > 0; s >>= 1) {
    if (tid < s) red[tid] += red[tid + s];
    __syncthreads();
  }
  if (tid == 0) partial[j] = red[0];
}

// Final reduction of 256 per-column partials by ONE wave32 using FP32 WMMA.
// A = ones(16x4). For each of 4 chunks of 64 values fed as B (4x16),
// D accumulates chunk column-sums replicated across all 16 rows. Therefore
// sum(all 256 entries of D) == 16 * sum(all 256 partials), independent of the
// exact B operand lane/VGPR layout. Exact FP32 math (RNE) -> deterministic.
__global__ __launch_bounds__(32) void final_reduce_wmma(
    const float* __restrict__ partial,
    float* __restrict__ out) {
  const int lane = threadIdx.x;  // 0..31, one full wave => EXEC all ones

  // Each lane pulls 8 of the 256 partials (coalesced).
  float pv[8];
#pragma unroll
  for (int t = 0; t < 8; ++t) pv[t] = partial[t * 32 + lane];

  const v2f a = {1.0f, 1.0f};    // all-ones A matrix (16x4 f32 = 2 VGPRs)
  v8f c = {};                    // 16x16 f32 accumulator (8 VGPRs)
#pragma unroll
  for (int s = 0; s < 4; ++s) {
    v2f b;
    b.x = pv[2 * s];
    b.y = pv[2 * s + 1];
    // 8 args: (neg_a, A, neg_b, B, c_mod, C, reuse_a, reuse_b)
    c = __builtin_amdgcn_wmma_f32_16x16x4_f32(
        /*neg_a=*/false, a, /*neg_b=*/false, b,
        /*c_mod=*/(short)0, c, /*reuse_a=*/false, /*reuse_b=*/false);
  }

  // Per-lane sum of the 8 accumulator registers, then 32-lane LDS finish.
  float acc = 0.0f;
#pragma unroll
  for (int r = 0; r < 8; ++r) acc += c[r];

  __shared__ float tmp[32];
  tmp[lane] = acc;
  __syncthreads();
  if (lane == 0) {
    float s = 0.0f;
    for (int n = 0; n < 32; ++n) s += tmp[n];
    // s == 16 * sum(partials); result = (sum m_r + sum m_l) / (H*W)
    out[0] = s * (1.0f / 16.0f) * (1.0f / (float)(H * W));
  }
}

extern "C" void kernel_launch(void* const* d_in, const int* in_sizes, int n_in,
                              void* d_out, int out_size, void* d_ws, size_t ws_size,
                              hipStream_t stream) {
  const float* up    = (const float*)d_in[0];
  const float* left  = (const float*)d_in[1];
  const float* right = (const float*)d_in[2];
  float* out = (float*)d_out;
  float* partial = (float*)d_ws;   // 256 floats = 1 KB of scratch

  scatter_loss_kernel<<<W, 256, 0, stream>>>(up, left, right, partial);
  final_reduce_wmma<<<1, 32, 0, stream>>>(partial, out);
}